// GraphNet_25941602468495
// MI455X (gfx1250) — compile-verified
//
#include <hip/hip_runtime.h>
#include <hip/hip_bf16.h>
#include <math.h>

typedef __attribute__((ext_vector_type(2))) float v2f;
typedef __attribute__((ext_vector_type(8))) float v8f;

#define BN_EPS 1e-5f

// ---------------------------------------------------------------------------
// Degree count: deg[dst] += 1 for every real edge (self-loop added in dinv).
// ---------------------------------------------------------------------------
__global__ void gcn_degree_kernel(const int* __restrict__ dst2,
                                  float* __restrict__ deg, int nE) {
    int e = blockIdx.x * blockDim.x + threadIdx.x;
    if (e < nE) {
        atomicAdd(&deg[dst2[e << 1]], 1.0f);   // low word of int64 index
    }
}

__global__ void gcn_dinv_kernel(const float* __restrict__ deg,
                                float* __restrict__ dinv, int n) {
    int i = blockIdx.x * blockDim.x + threadIdx.x;
    if (i < n) {
        // self-loop -> deg >= 1 always, matches reference rsqrt(max(deg,1))
        dinv[i] = rsqrtf(deg[i] + 1.0f);
    }
}

// ---------------------------------------------------------------------------
// WMMA fp32 GEMM: out[M,NC] = A[M,K] @ W[K,NC], row-major, M mult of 16.
// Each wave computes one 16xNC row stripe: a single A fragment per k-step
// feeds NT = NC/16 WMMAs (A-reuse + ILP across NT accumulator chains).
// W is pre-swizzled into LDS in *fragment order*: for k-step kq and tile t,
// lane L's (b.x, b.y) pair lives at Wsv[(kq*NT + t)*32 + L]. The inner loop
// is then a single aligned ds_load_b64 per fragment (no register repacking),
// with lane-consecutive 8B addresses covering all 64 banks conflict-free.
// ---------------------------------------------------------------------------
template <int K, int NC>
__global__ __launch_bounds__(256) void gcn_gemm_wmma_f32(
    const float* __restrict__ A, const float* __restrict__ W,
    float* __restrict__ out, int M) {
    constexpr int NT = NC / 16;
    constexpr int KQ = K / 4;
    __shared__ v2f Wsv[KQ * NT * 32];

    // cooperative swizzle of W[K][NC] into fragment order
    for (int pi = threadIdx.x; pi < KQ * NT * 32; pi += 256) {
        const int lane = pi & 31;
        const int tmp  = pi >> 5;            // kq*NT + t
        const int t    = tmp % NT;
        const int kq   = tmp / NT;
        const int half = lane >> 4;
        const int l16  = lane & 15;
        const int krow = kq * 4 + 2 * half;
        const int col  = t * 16 + l16;
        v2f p;
        p.x = W[(krow + 0) * NC + col];
        p.y = W[(krow + 1) * NC + col];
        Wsv[pi] = p;
    }
    __syncthreads();

    const int lane = threadIdx.x & 31;
    const int wave = threadIdx.x >> 5;
    const int mt   = blockIdx.x * 8 + wave;
    if (mt * 16 >= M) return;            // wave-uniform; EXEC all-1 for WMMA

    const int half = lane >> 4;          // 0: lanes 0-15, 1: lanes 16-31
    const int l16  = lane & 15;
    const float* __restrict__ arow = A + (long)(mt * 16 + l16) * K;

    v8f acc[NT];
#pragma unroll
    for (int t = 0; t < NT; ++t) acc[t] = (v8f){};

#pragma unroll 4
    for (int kq = 0; kq < KQ; ++kq) {
        // A fragment 16x4 f32: VGPR0 = K=(0|2), VGPR1 = K=(1|3) by lane half
        const int ka = kq * 4 + 2 * half;
        v2f a;
        a.x = arow[ka + 0];
        a.y = arow[ka + 1];
#pragma unroll
        for (int t = 0; t < NT; ++t) {
            // B fragment 4x16 f32: one aligned ds_load_b64 per fragment
            v2f b = Wsv[(kq * NT + t) * 32 + lane];
            acc[t] = __builtin_amdgcn_wmma_f32_16x16x4_f32(
                /*neg_a=*/false, a, /*neg_b=*/false, b,
                /*c_mod=*/(short)0, acc[t], /*reuse_a=*/false, /*reuse_b=*/false);
        }
    }

    // C/D 16x16 f32 layout: VGPR r -> M = r + 8*half, N = lane&15
#pragma unroll
    for (int t = 0; t < NT; ++t) {
#pragma unroll
        for (int r = 0; r < 8; ++r) {
            out[(long)(mt * 16 + r + 8 * half) * NC + t * 16 + l16] = acc[t][r];
        }
    }
}

// ---------------------------------------------------------------------------
// Edge scatter: agg[dst] += H[src] * dinv[src]*dinv[dst].
// One thread per (edge, float4 chunk); gathers are 16B coalesced per row.
// width is in floats (64 or 16), multiple of 4. Edge indices read as the low
// 32-bit word of the int64 (valid: 0 <= idx < N).
// ---------------------------------------------------------------------------
__global__ void gcn_scatter_kernel(const float* __restrict__ H,
                                   const int* __restrict__ src2,
                                   const int* __restrict__ dst2,
                                   const float* __restrict__ dinv,
                                   float* __restrict__ agg,
                                   int nE, int width) {
    const int wq  = width >> 2;                      // float4 chunks per row
    int idx = blockIdx.x * blockDim.x + threadIdx.x;
    int e = idx / wq;
    int q = idx - e * wq;
    if (e >= nE) return;

    const int s = src2[e << 1];
    const int d = dst2[e << 1];
    const float w = dinv[s] * dinv[d];

    const float4 h = ((const float4*)(H + (long)s * width))[q];
    float* __restrict__ o = agg + (long)d * width + q * 4;
    atomicAdd(o + 0, h.x * w);
    atomicAdd(o + 1, h.y * w);
    atomicAdd(o + 2, h.z * w);
    atomicAdd(o + 3, h.w * w);
}

// ---------------------------------------------------------------------------
// Fused epilogue (64-wide): self-loop + bias + BatchNorm(eval) + ReLU.
// Safe in-place on agg (pure elementwise).
// ---------------------------------------------------------------------------
__global__ void gcn_bn_relu_kernel(const float* __restrict__ H,
                                   const float* __restrict__ agg,
                                   const float* __restrict__ dinv,
                                   const float* __restrict__ bias,
                                   const float* __restrict__ g,
                                   const float* __restrict__ be,
                                   const float* __restrict__ m,
                                   const float* __restrict__ v,
                                   float* __restrict__ out, int n) {
    int idx = blockIdx.x * blockDim.x + threadIdx.x;
    if (idx >= n * 64) return;
    const int i = idx >> 6;
    const int j = idx & 63;
    const float di = dinv[i];
    float val = agg[idx] + H[idx] * di * di + bias[j];
    val = g[j] * (val - m[j]) * rsqrtf(v[j] + BN_EPS) + be[j];
    out[idx] = fmaxf(val, 0.0f);
}

// ---------------------------------------------------------------------------
// Final epilogue (16-wide): self-loop + bias + sigmoid -> d_out (float32).
// ---------------------------------------------------------------------------
__global__ void gcn_final_kernel(const float* __restrict__ H,
                                 const float* __restrict__ agg,
                                 const float* __restrict__ dinv,
                                 const float* __restrict__ bias,
                                 float* __restrict__ out, int n) {
    int idx = blockIdx.x * blockDim.x + threadIdx.x;
    if (idx >= n * 16) return;
    const int i = idx >> 4;
    const int j = idx & 15;
    const float di = dinv[i];
    const float val = agg[idx] + H[idx] * di * di + bias[j];
    out[idx] = 1.0f / (1.0f + __expf(-val));
}

// ---------------------------------------------------------------------------
extern "C" void kernel_launch(void* const* d_in, const int* in_sizes, int n_in,
                              void* d_out, int out_size, void* d_ws, size_t ws_size,
                              hipStream_t stream) {
    const float* x  = (const float*)d_in[0];
    const int*   ei = (const int*)d_in[1];   // int64 pairs, low words at 2*e
    const float* W1 = (const float*)d_in[2];
    const float* b1 = (const float*)d_in[3];
    const float* W2 = (const float*)d_in[4];
    const float* b2 = (const float*)d_in[5];
    const float* W3 = (const float*)d_in[6];
    const float* b3 = (const float*)d_in[7];
    const float* g1  = (const float*)d_in[8];
    const float* be1 = (const float*)d_in[9];
    const float* m1  = (const float*)d_in[10];
    const float* v1  = (const float*)d_in[11];
    const float* g2  = (const float*)d_in[12];
    const float* be2 = (const float*)d_in[13];
    const float* m2  = (const float*)d_in[14];
    const float* v2  = (const float*)d_in[15];

    const int N = in_sizes[0] / 128;    // 100000
    const int E = in_sizes[1] / 2;      // 1600000
    const int* src2 = ei;               // src int64 array viewed as int32 pairs
    const int* dst2 = ei + 2L * E;      // dst int64 array viewed as int32 pairs

    // workspace layout (floats)
    float* ws   = (float*)d_ws;
    float* deg  = ws;                    // N
    float* dinv = ws + N;                // N
    float* bufA = ws + 2L * N;           // N*64
    float* bufB = bufA + 64L * N;        // N*64
    float* bufC = bufB + 64L * N;        // N*64

    // --- graph normalization ---
    hipMemsetAsync(deg, 0, (size_t)N * sizeof(float), stream);
    gcn_degree_kernel<<<(E + 255) / 256, 256, 0, stream>>>(dst2, deg, E);
    gcn_dinv_kernel<<<(N + 255) / 256, 256, 0, stream>>>(deg, dinv, N);

    const int mblocks = (N / 16 + 7) / 8;

    // --- layer 1: H1 = X @ W1 (K=128, NC=64) ---
    gcn_gemm_wmma_f32<128, 64><<<mblocks, 256, 0, stream>>>(x, W1, bufA, N);
    hipMemsetAsync(bufB, 0, (size_t)N * 64 * sizeof(float), stream);
    gcn_scatter_kernel<<<((long)E * 16 + 255) / 256, 256, 0, stream>>>(
        bufA, src2, dst2, dinv, bufB, E, 64);
    gcn_bn_relu_kernel<<<(N * 64 + 255) / 256, 256, 0, stream>>>(
        bufA, bufB, dinv, b1, g1, be1, m1, v1, bufB, N);   // act1 -> bufB

    // --- layer 2: H2 = act1 @ W2 (K=64, NC=64) ---
    gcn_gemm_wmma_f32<64, 64><<<mblocks, 256, 0, stream>>>(bufB, W2, bufA, N);
    hipMemsetAsync(bufC, 0, (size_t)N * 64 * sizeof(float), stream);
    gcn_scatter_kernel<<<((long)E * 16 + 255) / 256, 256, 0, stream>>>(
        bufA, src2, dst2, dinv, bufC, E, 64);
    gcn_bn_relu_kernel<<<(N * 64 + 255) / 256, 256, 0, stream>>>(
        bufA, bufC, dinv, b2, g2, be2, m2, v2, bufC, N);   // act2 -> bufC

    // --- layer 3: H3 = act2 @ W3 (K=64, NC=16) ---
    float* H3   = bufA;                  // N*16 (reuse)
    float* agg3 = bufA + 16L * N;        // N*16 (reuse)
    gcn_gemm_wmma_f32<64, 16><<<mblocks, 256, 0, stream>>>(bufC, W3, H3, N);
    hipMemsetAsync(agg3, 0, (size_t)N * 16 * sizeof(float), stream);
    gcn_scatter_kernel<<<((long)E * 4 + 255) / 256, 256, 0, stream>>>(
        H3, src2, dst2, dinv, agg3, E, 16);
    gcn_final_kernel<<<(N * 16 + 255) / 256, 256, 0, stream>>>(
        H3, agg3, dinv, b3, (float*)d_out, N);
}